// WSCNet_63359357551231
// MI455X (gfx1250) — compile-verified
//
#include <hip/hip_runtime.h>
#include <hip/hip_bf16.h>

typedef __attribute__((ext_vector_type(8))) float v8f;
typedef __attribute__((ext_vector_type(2))) float v2f;

#define B_DIM 64
#define C_DIM 2048
#define HW 196
#define O_DIM 32
// 64*196 = 12544 pixels = 784 tiles of 16 = 196 waves x 4 tiles (exact, no tail)

// ---------------- Kernel 1: transpose conv_w (32x2048) -> wT (2048x32) ----------------
__global__ __launch_bounds__(256) void wsc_transpose_w(const float* __restrict__ cw,
                                                       float* __restrict__ wT) {
    int idx = blockIdx.x * blockDim.x + threadIdx.x;   // 65536 total
    if (idx < O_DIM * C_DIM) {
        int o = idx / C_DIM, c = idx % C_DIM;          // coalesced read
        wT[c * O_DIM + o] = cw[idx];
    }
}

// ---------------- Kernel 2: 1x1 conv via V_WMMA_F32_16X16X4_F32 ----------------
// One wave owns 4 pixel-tiles (64 global pixels) x 32 outputs; the B-fragment
// pair is loaded once per K-step and reused by 4 WMMA pairs. Pixel rows are
// per-lane, so (b,hw) is derived per lane from a flat pixel id -> no padding.
__global__ __launch_bounds__(128) void wsc_conv_wmma(const float* __restrict__ feat,
                                                     const float* __restrict__ wT,
                                                     const float* __restrict__ conv_bias,
                                                     float* __restrict__ x_ws) {
    int wave = (blockIdx.x * blockDim.x + threadIdx.x) >> 5;   // 0..195
    int lane = threadIdx.x & 31;
    int half = lane >> 4;          // 0: K={0,1}, 1: K={2,3}
    int l15  = lane & 15;          // A: M row; B: N col
    int kk0  = half * 2;

    // Per-tile, per-lane A base offsets (flat pixel -> (b, hw))
    int pgBase = wave * 64;
    size_t abase[4];
    #pragma unroll
    for (int i = 0; i < 4; ++i) {
        int pg = pgBase + i * 16 + l15;
        int b  = pg / HW;
        int hw = pg - b * HW;
        abase[i] = (size_t)b * C_DIM * HW + (size_t)hw;
    }

    v8f acc[4][2];
    #pragma unroll
    for (int i = 0; i < 4; ++i) {
        acc[i][0] = (v8f){0.f,0.f,0.f,0.f,0.f,0.f,0.f,0.f};
        acc[i][1] = (v8f){0.f,0.f,0.f,0.f,0.f,0.f,0.f,0.f};
    }

    #pragma unroll 2
    for (int c = 0; c < C_DIM; c += 4) {
        // B-matrix 4x16 f32: lane = N, v0/v1 = K pair (wT is [c][o], coalesced)
        const float* w0 = wT + (size_t)(c + kk0) * O_DIM;
        const float* w1 = w0 + O_DIM;
        v2f b0, b1;
        b0.x = w0[l15];       b0.y = w1[l15];        // outputs 0..15
        b1.x = w0[l15 + 16];  b1.y = w1[l15 + 16];   // outputs 16..31

        size_t koff0 = (size_t)(c + kk0) * HW;
        size_t koff1 = koff0 + HW;
        #pragma unroll
        for (int i = 0; i < 4; ++i) {
            v2f a;   // A-matrix 16x4 f32: v0 = K=kk0, v1 = K=kk0+1, M = l15
            a.x = feat[abase[i] + koff0];
            a.y = feat[abase[i] + koff1];
            acc[i][0] = __builtin_amdgcn_wmma_f32_16x16x4_f32(false, a, false, b0,
                                                              (short)0, acc[i][0], false, false);
            acc[i][1] = __builtin_amdgcn_wmma_f32_16x16x4_f32(false, a, false, b1,
                                                              (short)0, acc[i][1], false, false);
        }
    }

    // C/D layout: lane l15 = N (output chan), vgpr r -> M = r + 8*half (pixel)
    float bias0 = conv_bias[l15];
    float bias1 = conv_bias[l15 + 16];
    #pragma unroll
    for (int i = 0; i < 4; ++i) {
        #pragma unroll
        for (int r = 0; r < 8; ++r) {
            int pg = pgBase + i * 16 + r + half * 8;
            int b  = pg / HW;
            int hw = pg - b * HW;
            float* xb = x_ws + (size_t)b * O_DIM * HW;
            xb[(size_t)l15        * HW + hw] = acc[i][0][r] + bias0;
            xb[(size_t)(l15 + 16) * HW + hw] = acc[i][1][r] + bias1;
        }
    }
}

// ---------------- Kernel 3: per-batch gmp -> xv (output 1) -> spatial map s ----------------
__global__ __launch_bounds__(256) void wsc_stats(const float* __restrict__ x_ws,
                                                 float* __restrict__ xv_out,
                                                 float* __restrict__ s_ws) {
    int b = blockIdx.x;
    __shared__ float red[O_DIM * 8];
    __shared__ float gmp[O_DIM];
    __shared__ float xv[8];
    const float* xb = x_ws + (size_t)b * O_DIM * HW;
    int t = threadIdx.x;

    int o = t >> 3, j = t & 7;                 // 32 outputs x 8 partials
    float mx = xb[o * HW + j];                 // j < 8 <= 196, always valid
    for (int p = j + 8; p < HW; p += 8) mx = fmaxf(mx, xb[o * HW + p]);
    red[t] = mx;
    __syncthreads();

    if (t < O_DIM) {
        float m = red[t * 8];
        #pragma unroll
        for (int q = 1; q < 8; ++q) m = fmaxf(m, red[t * 8 + q]);
        gmp[t] = m;
    }
    __syncthreads();

    if (t < 8) {
        float v = 0.25f * (gmp[4*t] + gmp[4*t+1] + gmp[4*t+2] + gmp[4*t+3]);
        xv[t] = v;
        xv_out[b * 8 + t] = v;                 // first tuple output
    }
    __syncthreads();

    if (t < HW) {
        float s = 0.f;
        #pragma unroll
        for (int o2 = 0; o2 < O_DIM; ++o2) s += xv[o2 >> 2] * xb[o2 * HW + t];
        s_ws[b * HW + t] = s * (1.0f / 32.0f);
    }
}

// ---------------- Kernel 4: second pass over features -> gap (b,4096) ----------------
// block = (b, chunk of 8 channels); 8 waves, each wave reduces one 196-row twice.
// Rows are 49 aligned float4s -> b128 loads.
__global__ __launch_bounds__(256) void wsc_gap(const float* __restrict__ feat,
                                               const float* __restrict__ s_ws,
                                               float* __restrict__ gap_ws) {
    int b     = blockIdx.x >> 8;        // 256 chunks of 8 channels
    int chunk = blockIdx.x & 255;
    __shared__ float4 s_sh4[HW / 4];    // 49
    int t = threadIdx.x;
    if (t < HW / 4) s_sh4[t] = ((const float4*)(s_ws + (size_t)b * HW))[t];
    __syncthreads();

    int wave = t >> 5, lane = t & 31;
    int c = chunk * 8 + wave;
    const float4* row4 = (const float4*)(feat + ((size_t)b * C_DIM + c) * HW);
    float s1 = 0.f, s2 = 0.f;
    for (int p = lane; p < HW / 4; p += 32) {
        float4 f  = row4[p];
        float4 sv = s_sh4[p];
        s1 += (f.x + f.y) + (f.z + f.w);
        s2 += (f.x * sv.x + f.y * sv.y) + (f.z * sv.z + f.w * sv.w);
    }
    #pragma unroll
    for (int off = 16; off; off >>= 1) {
        s1 += __shfl_xor(s1, off, 32);
        s2 += __shfl_xor(s2, off, 32);
    }
    if (lane == 0) {
        gap_ws[(size_t)b * 4096 + c]        = s1 * (1.0f / (float)HW);
        gap_ws[(size_t)b * 4096 + 2048 + c] = s2 * (1.0f / (float)HW);
    }
}

// ---------------- Kernel 5: final FC -> res (output 2) ----------------
__global__ __launch_bounds__(256) void wsc_fc(const float* __restrict__ gap_ws,
                                              const float* __restrict__ fc_w,
                                              const float* __restrict__ fc_b,
                                              float* __restrict__ res_out) {
    int wave = (blockIdx.x * blockDim.x + threadIdx.x) >> 5;  // 512 waves
    int lane = threadIdx.x & 31;
    int b = wave >> 3, k = wave & 7;
    const float4* g = (const float4*)(gap_ws + (size_t)b * 4096);
    const float4* w = (const float4*)(fc_w   + (size_t)k * 4096);
    float s = 0.f;
    for (int j = lane; j < 1024; j += 32) {
        float4 gv = g[j];
        float4 wv = w[j];
        s += (gv.x * wv.x + gv.y * wv.y) + (gv.z * wv.z + gv.w * wv.w);
    }
    #pragma unroll
    for (int off = 16; off; off >>= 1) s += __shfl_xor(s, off, 32);
    if (lane == 0) res_out[b * 8 + k] = s + fc_b[k];
}

extern "C" void kernel_launch(void* const* d_in, const int* in_sizes, int n_in,
                              void* d_out, int out_size, void* d_ws, size_t ws_size,
                              hipStream_t stream) {
    const float* feat    = (const float*)d_in[0];   // (64,2048,14,14)
    const float* conv_w  = (const float*)d_in[1];   // (32,2048)
    const float* conv_b  = (const float*)d_in[2];   // (32,)
    const float* fc_w    = (const float*)d_in[3];   // (8,4096)
    const float* fc_b    = (const float*)d_in[4];   // (8,)
    float* out = (float*)d_out;                     // [xv (64*8) | res (64*8)]

    char* ws = (char*)d_ws;
    float* wT     = (float*)(ws);                                   // 2048*32 f32 = 256 KB
    float* x_ws   = (float*)(ws + 262144);                          // 64*32*196  = 1.53 MB
    float* s_ws   = (float*)(ws + 262144 + 1605632);                // 64*196
    float* gap_ws = (float*)(ws + 262144 + 1605632 + 50176);        // 64*4096    = 1 MB

    wsc_transpose_w<<<256, 256, 0, stream>>>(conv_w, wT);
    // 784 tiles / 4 per wave = 196 waves = 49 blocks x 4 waves (128 threads)
    wsc_conv_wmma<<<49, 128, 0, stream>>>(feat, wT, conv_b, x_ws);
    wsc_stats<<<B_DIM, 256, 0, stream>>>(x_ws, out, s_ws);
    wsc_gap<<<B_DIM * 256, 256, 0, stream>>>(feat, s_ws, gap_ws);
    wsc_fc<<<64, 256, 0, stream>>>(gap_ws, fc_w, fc_b, out + 512);
}